// HeteroRGCN_35648228556927
// MI455X (gfx1250) — compile-verified
//
#include <hip/hip_runtime.h>
#include <hip/hip_bf16.h>

// ---------------------------------------------------------------------------
// HeteroRGCN layer on MI455X (gfx1250, wave32):
//   Wh_likes   = user_feats  @ W_likes   + b    (WMMA f32 16x16x4, W staged
//                                               into LDS via async-to-LDS DMA)
//   Wh_likedby = photo_feats @ W_likedby + b
//   h_photo    = mean-scatter(Wh_likes[likes_src]     -> likes_dst)
//   h_user     = mean-scatter(Wh_likedby[likedby_src] -> likedby_dst)
// ---------------------------------------------------------------------------

#define N_USER  100000
#define N_PHOTO 200000
#define NEDGE   1000000
#define KIN     256
#define NOUT    64

typedef float v2f __attribute__((ext_vector_type(2)));
typedef float v8f __attribute__((ext_vector_type(8)));

// ---------------------------------------------------------------------------
// Zero-fill (grid-stride) — d_out / degree arrays must start at 0 each call.
// ---------------------------------------------------------------------------
__global__ void zero_f32(float* __restrict__ p, size_t n) {
    size_t i = (size_t)blockIdx.x * blockDim.x + threadIdx.x;
    size_t stride = (size_t)gridDim.x * blockDim.x;
    for (; i < n; i += stride) p[i] = 0.0f;
}

// ---------------------------------------------------------------------------
// GEMM + bias: Out[M x 64] = A[M x 256] * W[256 x 64] + bias
//
// W (64 KB) is staged once per workgroup into LDS with the gfx1250 async
// copy path (GLOBAL_LOAD_ASYNC_TO_LDS_B128, ASYNCcnt), so the inner-loop
// vmem pipe carries only the A stream (the true HBM traffic).
//
// One wave32 owns a 16-row tile; 4 accumulator tiles span N=64.
// K-loop: 64 iterations of V_WMMA_F32_16X16X4_F32.
//
// Fragment layouts (CDNA5 ISA 7.12.2, 32-bit, wave32):
//   A 16x4 : lane<16 -> {K=0,K=1}, lane>=16 -> {K=2,K=3}, row M = lane&15
//   B 4x16 : mirrored, col N = lane&15
//   C/D    : VGPR r -> row (r + 8*(lane>>4)), col (lane&15)
// ---------------------------------------------------------------------------
__global__ __launch_bounds__(256) void gemm_bias_wmma(
    const float* __restrict__ A,
    const float* __restrict__ W,
    const float* __restrict__ bias,
    float* __restrict__ Out,
    int M)
{
    __shared__ float Wlds[KIN * NOUT];   // 64 KB of the WGP's 320 KB

    // ---- cooperative async stage of W into LDS (all threads, incl. waves
    //      that will early-exit: loop covers all 4096 16-byte chunks) -------
    for (int i = threadIdx.x; i < (KIN * NOUT) / 4; i += blockDim.x) {
        const float* g = W + (size_t)i * 4;
        unsigned loff = (unsigned)(uintptr_t)(&Wlds[i * 4]);  // LDS offset = addr[31:0]
        asm volatile("global_load_async_to_lds_b128 %0, %1, off"
                     :: "v"(loff), "v"(g) : "memory");
    }
#if __has_builtin(__builtin_amdgcn_s_wait_asynccnt)
    __builtin_amdgcn_s_wait_asynccnt(0);
#else
    asm volatile("s_wait_asynccnt 0x0" ::: "memory");
#endif
    __syncthreads();

    const int lane  = threadIdx.x & 31;
    const int wave  = blockIdx.x * (blockDim.x >> 5) + (threadIdx.x >> 5);
    const int m0    = wave << 4;
    if (m0 >= M) return;                 // wave-uniform: EXEC stays all-ones

    const int halfK = (lane >> 4) << 1;  // 0 or 2
    const int l15   = lane & 15;

    v8f acc0 = {}, acc1 = {}, acc2 = {}, acc3 = {};

    const float* arow = A + (size_t)(m0 + l15) * KIN + halfK;

    for (int k = 0; k < KIN; k += 4) {
        // A fragment: two consecutive f32 (8B-aligned vector load, HBM stream)
        v2f a = *(const v2f*)(arow + k);

        // B fragments from LDS: rows k+halfK, k+halfK+1; col = tile*16 + l15
        const float* brow = &Wlds[(k + halfK) * NOUT + l15];
        v2f b0, b1, b2, b3;
        b0.x = brow[0];    b0.y = brow[NOUT + 0];
        b1.x = brow[16];   b1.y = brow[NOUT + 16];
        b2.x = brow[32];   b2.y = brow[NOUT + 32];
        b3.x = brow[48];   b3.y = brow[NOUT + 48];

        acc0 = __builtin_amdgcn_wmma_f32_16x16x4_f32(false, a, false, b0, (short)0, acc0, false, false);
        acc1 = __builtin_amdgcn_wmma_f32_16x16x4_f32(false, a, false, b1, (short)0, acc1, false, false);
        acc2 = __builtin_amdgcn_wmma_f32_16x16x4_f32(false, a, false, b2, (short)0, acc2, false, false);
        acc3 = __builtin_amdgcn_wmma_f32_16x16x4_f32(false, a, false, b3, (short)0, acc3, false, false);
    }

    // Store with bias: lane's VGPR r holds row (m0 + r + 8*(lane>>4)), col l15+16j
    const int rbase = m0 + ((lane >> 4) << 3);
    v8f accs[4] = {acc0, acc1, acc2, acc3};
#pragma unroll
    for (int j = 0; j < 4; ++j) {
        const int col = (j << 4) + l15;
        const float bv = bias[col];
        float* orow = Out + (size_t)rbase * NOUT + col;
#pragma unroll
        for (int r = 0; r < 8; ++r)
            orow[(size_t)r * NOUT] = accs[j][r] + bv;
    }
}

// ---------------------------------------------------------------------------
// Edge scatter: 16 lanes per edge, float4 per lane.
// out_accum and deg must be pre-zeroed. Native f32 global atomics (no-return
// -> STOREcnt path).
// ---------------------------------------------------------------------------
__global__ __launch_bounds__(256) void scatter_accum(
    const float* __restrict__ Wh,     // [n_src x 64]
    const int*   __restrict__ src,
    const int*   __restrict__ dst,
    float* __restrict__ out_accum,    // [n_dst x 64]
    float* __restrict__ deg,          // [n_dst]
    int nedge)
{
    const int gid = blockIdx.x * blockDim.x + threadIdx.x;
    const int e   = gid >> 4;
    if (e >= nedge) return;
    const int c4  = (gid & 15) << 2;

    const int s = src[e];
    const int d = dst[e];

    const float4 v = *(const float4*)(Wh + (size_t)s * NOUT + c4);
    float* o = out_accum + (size_t)d * NOUT + c4;
    unsafeAtomicAdd(o + 0, v.x);
    unsafeAtomicAdd(o + 1, v.y);
    unsafeAtomicAdd(o + 2, v.z);
    unsafeAtomicAdd(o + 3, v.w);
    if (c4 == 0) unsafeAtomicAdd(deg + d, 1.0f);
}

// ---------------------------------------------------------------------------
// Mean: divide each node row by max(degree, 1)
// ---------------------------------------------------------------------------
__global__ __launch_bounds__(256) void normalize_by_deg(
    float* __restrict__ h, const float* __restrict__ deg, int nnode)
{
    const int gid = blockIdx.x * blockDim.x + threadIdx.x;
    if (gid >= nnode * NOUT) return;
    const float dv = deg[gid >> 6];
    h[gid] = h[gid] / fmaxf(dv, 1.0f);
}

// ---------------------------------------------------------------------------
extern "C" void kernel_launch(void* const* d_in, const int* in_sizes, int n_in,
                              void* d_out, int out_size, void* d_ws, size_t ws_size,
                              hipStream_t stream) {
    const float* user_feats  = (const float*)d_in[0];
    const float* photo_feats = (const float*)d_in[1];
    const float* W_likes     = (const float*)d_in[2];
    const float* b_likes     = (const float*)d_in[3];
    const float* W_likedby   = (const float*)d_in[4];
    const float* b_likedby   = (const float*)d_in[5];
    const int* likes_src     = (const int*)d_in[6];
    const int* likes_dst     = (const int*)d_in[7];
    const int* likedby_src   = (const int*)d_in[8];
    const int* likedby_dst   = (const int*)d_in[9];

    float* out     = (float*)d_out;
    float* h_user  = out;                              // [N_USER  x 64]
    float* h_photo = out + (size_t)N_USER * NOUT;      // [N_PHOTO x 64]

    float* ws         = (float*)d_ws;
    float* Wh_likes   = ws;                                    // [N_USER  x 64]
    float* Wh_likedby = Wh_likes + (size_t)N_USER * NOUT;      // [N_PHOTO x 64]
    float* deg_photo  = Wh_likedby + (size_t)N_PHOTO * NOUT;   // [N_PHOTO]
    float* deg_user   = deg_photo + N_PHOTO;                   // [N_USER]

    // 1) zero outputs + degree counters
    zero_f32<<<2048, 256, 0, stream>>>(out, (size_t)(N_USER + N_PHOTO) * NOUT);
    zero_f32<<<512, 256, 0, stream>>>(deg_photo, (size_t)(N_PHOTO + N_USER));

    // 2) per-etype projections (WMMA GEMMs)
    const int tiles_u = N_USER / 16;    // 6250
    const int tiles_p = N_PHOTO / 16;   // 12500
    gemm_bias_wmma<<<(tiles_u + 7) / 8, 256, 0, stream>>>(
        user_feats, W_likes, b_likes, Wh_likes, N_USER);
    gemm_bias_wmma<<<(tiles_p + 7) / 8, 256, 0, stream>>>(
        photo_feats, W_likedby, b_likedby, Wh_likedby, N_PHOTO);

    // 3) edge scatter-sum (+degree)
    const int sblocks = (NEDGE * 16) / 256;   // 62500
    scatter_accum<<<sblocks, 256, 0, stream>>>(
        Wh_likes, likes_src, likes_dst, h_photo, deg_photo, NEDGE);
    scatter_accum<<<sblocks, 256, 0, stream>>>(
        Wh_likedby, likedby_src, likedby_dst, h_user, deg_user, NEDGE);

    // 4) mean
    normalize_by_deg<<<(N_PHOTO * NOUT + 255) / 256, 256, 0, stream>>>(
        h_photo, deg_photo, N_PHOTO);
    normalize_by_deg<<<(N_USER * NOUT + 255) / 256, 256, 0, stream>>>(
        h_user, deg_user, N_USER);
}